// InvariantPolynomial_1486058684778
// MI455X (gfx1250) — compile-verified
//
#include <hip/hip_runtime.h>
#include <math.h>

typedef __attribute__((ext_vector_type(2))) float v2f;
typedef __attribute__((ext_vector_type(8))) float v8f;

// global-address-space pointer to v2f: loads lower to global_load_b64 (LOADcnt only,
// no DScnt coupling with the LDS A-operand stream)
typedef const v2f __attribute__((address_space(1)))* gv2p;

#define TROWS   96      // padded t-rows (86 used)
#define NCOLS   384     // padded mid dim (368 used)
#define NE1MAX  2048
#define NE2MAX  512
#define T2ROWS  144
#define INV_SQRT2F 0.70710678118654752440f

struct Ent { int r; int a; int b; float v; };

// ---------------- precomputed tables (rebuilt deterministically every launch) ----
// g_M1s: t[96] -> mid[384] dense weight matrix (incl 1/sqrt2), stored pre-swizzled in
// WMMA B-operand lane order: element (k, col) at [((kk*2+half)*NCOLS + col)*2 + r]
// with k = kk*4 + 2*half + r.  One global_load_b64 per lane per K-step.
__device__ float g_M1s[TROWS * NCOLS];
__device__ float g_M2[T2ROWS * 8];      // t2[144] -> out[7] (incl 0.5/sqrt2)
__device__ int   g_rowcnt[TROWS + 1];
__device__ int   g_row_start[TROWS + 1];
__device__ Ent   g_e1[NE1MAX];          // TP1 wigner nnz, sorted by t-row
__device__ Ent   g_e2[NE2MAX];          // TP2 wigner nnz, flat
__device__ int   g_p2cnt[8];
__device__ int   g_p2start[8];
__device__ int   g_ne2;

// irreps metadata (matches reference)
__device__ const int SH_L[4]   = {0, 1, 2, 3};
__device__ const int SH_P[4]   = {1, -1, 1, -1};
__device__ const int SH_OFF[4] = {0, 1, 4, 9};
__device__ const int MID_MUL[5] = {64, 24, 24, 16, 16};
__device__ const int MID_L[5]   = {0, 1, 1, 2, 2};
__device__ const int MID_P[5]   = {1, 1, -1, 1, -1};
__device__ const int MID_OFF[5] = {0, 64, 136, 208, 288};
__device__ const int OUT_MUL[2] = {1, 6};
__device__ const int OUT_L[2]   = {0, 0};
__device__ const int OUT_P[2]   = {-1, 1};
__device__ const int OUT_OFF[2] = {0, 1};

__device__ inline int iabs_i(int x) { return x < 0 ? -x : x; }

// swizzled index for logical M1 element (row=k, col)
__device__ inline int m1s_index(int row, int col) {
    int kk = row >> 2;
    int half = (row >> 1) & 1;
    int r = row & 1;
    return ((kk * 2 + half) * NCOLS + col) * 2 + r;
}

__device__ double dfact(int n) {
    double r = 1.0;
    for (int i = 2; i <= n; ++i) r *= (double)i;
    return r;
}

// <j1 m1 j2 m2 | j3 m3>, Racah formula (integer j)
__device__ double su2_cg(int j1, int j2, int j3, int m1, int m2, int m3) {
    if (m3 != m1 + m2) return 0.0;
    if (j3 < iabs_i(j1 - j2) || j3 > j1 + j2) return 0.0;
    double pref = sqrt((double)(2 * j3 + 1)
        * dfact(j1 + j2 - j3) * dfact(j1 - j2 + j3) * dfact(-j1 + j2 + j3) / dfact(j1 + j2 + j3 + 1)
        * dfact(j3 + m3) * dfact(j3 - m3) * dfact(j1 - m1) * dfact(j1 + m1)
        * dfact(j2 - m2) * dfact(j2 + m2));
    int kmin = 0;
    if (j2 - j3 - m1 > kmin) kmin = j2 - j3 - m1;
    if (j1 - j3 + m2 > kmin) kmin = j1 - j3 + m2;
    int kmax = j1 + j2 - j3;
    if (j1 - m1 < kmax) kmax = j1 - m1;
    if (j2 + m2 < kmax) kmax = j2 + m2;
    double s = 0.0;
    for (int k = kmin; k <= kmax; ++k) {
        double d = dfact(k) * dfact(j1 + j2 - j3 - k) * dfact(j1 - m1 - k)
                 * dfact(j2 + m2 - k) * dfact(j3 - j2 + m1 + k) * dfact(j3 - j1 - m2 + k);
        s += ((k & 1) ? -1.0 : 1.0) / d;
    }
    return pref * s;
}

// real->complex SH basis change (e3nn convention), q = (-i)^l * q0
__device__ void build_q(int l, double qR[7][7], double qI[7][7]) {
    for (int a = 0; a < 7; ++a)
        for (int b = 0; b < 7; ++b) { qR[a][b] = 0.0; qI[a][b] = 0.0; }
    const double is2 = 1.0 / sqrt(2.0);
    for (int m = -l; m < 0; ++m) {
        qR[l + m][l - m] = is2;     // q[l+m][l+|m|] = 1/sqrt2
        qI[l + m][l + m] = -is2;    // q[l+m][l-|m|] = -i/sqrt2
    }
    qR[l][l] = 1.0;
    for (int m = 1; m <= l; ++m) {
        double sg = (m & 1) ? -1.0 : 1.0;
        qR[l + m][l + m] = sg * is2;
        qI[l + m][l - m] = sg * is2;
    }
    double pr, pi;
    switch (l & 3) {
        case 0: pr = 1; pi = 0; break;
        case 1: pr = 0; pi = -1; break;
        case 2: pr = -1; pi = 0; break;
        default: pr = 0; pi = 1; break;
    }
    for (int a = 0; a < 7; ++a)
        for (int b = 0; b < 7; ++b) {
            double r = qR[a][b], im = qI[a][b];
            qR[a][b] = pr * r - pi * im;
            qI[a][b] = pr * im + pi * r;
        }
}

// real-basis Wigner 3j, unit Frobenius norm.  C[(i*7+j)*7+k]
__device__ void w3j_real(int l1, int l2, int l3, float* C) {
    double AR[7][7][7], AI[7][7][7], BR[7][7][7], BI[7][7][7];
    int d1 = 2 * l1 + 1, d2 = 2 * l2 + 1, d3 = 2 * l3 + 1;
    for (int i = 0; i < 7; ++i)
        for (int j = 0; j < 7; ++j)
            for (int k = 0; k < 7; ++k) { AR[i][j][k] = 0; AI[i][j][k] = 0; BR[i][j][k] = 0; BI[i][j][k] = 0; }
    for (int m1 = -l1; m1 <= l1; ++m1)
        for (int m2 = -l2; m2 <= l2; ++m2) {
            int m3 = m1 + m2;
            if (m3 < -l3 || m3 > l3) continue;
            AR[l1 + m1][l2 + m2][l3 + m3] = su2_cg(l1, l2, l3, m1, m2, m3);
        }
    double q1R[7][7], q1I[7][7], q2R[7][7], q2I[7][7], q3R[7][7], q3I[7][7];
    build_q(l1, q1R, q1I); build_q(l2, q2R, q2I); build_q(l3, q3R, q3I);
    // B[j][k][m] = sum_i q1[i][j]*A[i][k][m]
    for (int j = 0; j < d1; ++j)
        for (int k = 0; k < d2; ++k)
            for (int m = 0; m < d3; ++m) {
                double sr = 0, si = 0;
                for (int i = 0; i < d1; ++i) {
                    double ar = AR[i][k][m], ai = AI[i][k][m];
                    sr += q1R[i][j] * ar - q1I[i][j] * ai;
                    si += q1R[i][j] * ai + q1I[i][j] * ar;
                }
                BR[j][k][m] = sr; BI[j][k][m] = si;
            }
    // A[j][l][m] = sum_k q2[k][l]*B[j][k][m]
    for (int j = 0; j < d1; ++j)
        for (int l = 0; l < d2; ++l)
            for (int m = 0; m < d3; ++m) {
                double sr = 0, si = 0;
                for (int k = 0; k < d2; ++k) {
                    double br = BR[j][k][m], bi = BI[j][k][m];
                    sr += q2R[k][l] * br - q2I[k][l] * bi;
                    si += q2R[k][l] * bi + q2I[k][l] * br;
                }
                AR[j][l][m] = sr; AI[j][l][m] = si;
            }
    // B[j][l][n] = sum_m conj(q3)[m][n]*A[j][l][m]
    for (int j = 0; j < d1; ++j)
        for (int l = 0; l < d2; ++l)
            for (int n = 0; n < d3; ++n) {
                double sr = 0, si = 0;
                for (int m = 0; m < d3; ++m) {
                    double cr = q3R[m][n], ci = -q3I[m][n];
                    double ar = AR[j][l][m], ai = AI[j][l][m];
                    sr += cr * ar - ci * ai;
                    si += cr * ai + ci * ar;
                }
                BR[j][l][n] = sr; BI[j][l][n] = si;
            }
    double nr = 0, ni = 0;
    for (int j = 0; j < d1; ++j)
        for (int l = 0; l < d2; ++l)
            for (int n = 0; n < d3; ++n) { nr += BR[j][l][n] * BR[j][l][n]; ni += BI[j][l][n] * BI[j][l][n]; }
    bool useR = (nr >= ni);
    double inv = 1.0 / sqrt(useR ? nr : ni);
    for (int j = 0; j < d1; ++j)
        for (int l = 0; l < d2; ++l)
            for (int n = 0; n < d3; ++n)
                C[(j * 7 + l) * 7 + n] = (float)((useR ? BR[j][l][n] : BI[j][l][n]) * inv);
}

// ------------------------- setup: build all tables -------------------------
__global__ __launch_bounds__(256) void setup_kernel(const float* __restrict__ w1,
                                                    const float* __restrict__ w2) {
    const int tid = threadIdx.x;
    for (int i = tid; i < TROWS * NCOLS; i += 256) g_M1s[i] = 0.f;
    for (int i = tid; i < T2ROWS * 8; i += 256) g_M2[i] = 0.f;
    for (int i = tid; i <= TROWS; i += 256) g_rowcnt[i] = 0;
    if (tid < 8) g_p2cnt[tid] = 0;
    __syncthreads();

    float C[343];

    // ---- TP1: one thread per path (24 valid), rows exclusive per path ----
    int p1_i1 = -1, p1_i2 = 0, p1_i3 = 0, p1_woff = 0, p1_trow = 0;
    double p1_pw = 0.0;
    if (tid < 64) {
        int fan[5] = {0, 0, 0, 0, 0};
        for (int i1 = 0; i1 < 4; ++i1)
            for (int i2 = 0; i2 < 4; ++i2)
                for (int i3 = 0; i3 < 5; ++i3) {
                    int l1 = SH_L[i1], l2 = SH_L[i2], l3 = MID_L[i3];
                    if (SH_P[i1] * SH_P[i2] == MID_P[i3] && l3 >= iabs_i(l1 - l2) && l3 <= l1 + l2) fan[i3]++;
                }
        int idx = 0, woff = 0, trow = 0;
        for (int i1 = 0; i1 < 4; ++i1)
            for (int i2 = 0; i2 < 4; ++i2)
                for (int i3 = 0; i3 < 5; ++i3) {
                    int l1 = SH_L[i1], l2 = SH_L[i2], l3 = MID_L[i3];
                    if (!(SH_P[i1] * SH_P[i2] == MID_P[i3] && l3 >= iabs_i(l1 - l2) && l3 <= l1 + l2)) continue;
                    if (idx == tid) {
                        p1_i1 = i1; p1_i2 = i2; p1_i3 = i3; p1_woff = woff; p1_trow = trow;
                        p1_pw = sqrt((double)(2 * l3 + 1) / (double)fan[i3]);
                    }
                    woff += MID_MUL[i3];
                    trow += 2 * MID_L[i3] + 1;
                    ++idx;
                }
    }
    int p1_d1 = 0, p1_d2 = 0, p1_d3 = 0;
    if (p1_i1 >= 0) {
        int l1 = SH_L[p1_i1], l2 = SH_L[p1_i2], l3 = MID_L[p1_i3];
        p1_d1 = 2 * l1 + 1; p1_d2 = 2 * l2 + 1; p1_d3 = 2 * l3 + 1;
        w3j_real(l1, l2, l3, C);
        float scale = (float)p1_pw * INV_SQRT2F;   // fold second segment-sum scale
        int wcnt = MID_MUL[p1_i3];
        for (int w = 0; w < wcnt; ++w) {
            float wv = scale * w1[p1_woff + w];
            for (int k = 0; k < p1_d3; ++k)
                g_M1s[m1s_index(p1_trow + k, MID_OFF[p1_i3] + w * p1_d3 + k)] = wv;
        }
        for (int k = 0; k < p1_d3; ++k) {
            int cnt = 0;
            for (int i = 0; i < p1_d1; ++i)
                for (int j = 0; j < p1_d2; ++j)
                    if (fabsf(C[(i * 7 + j) * 7 + k]) > 1e-8f) ++cnt;
            g_rowcnt[p1_trow + k] = cnt;
        }
    }
    __syncthreads();
    if (tid == 0) {
        int acc = 0;
        for (int r = 0; r < TROWS; ++r) { g_row_start[r] = acc; acc += g_rowcnt[r]; }
        g_row_start[TROWS] = acc;
    }
    __syncthreads();
    if (p1_i1 >= 0) {
        for (int k = 0; k < p1_d3; ++k) {
            int p = g_row_start[p1_trow + k];
            for (int i = 0; i < p1_d1; ++i)
                for (int j = 0; j < p1_d2; ++j) {
                    float v = C[(i * 7 + j) * 7 + k];
                    if (fabsf(v) > 1e-8f) {
                        g_e1[p].r = p1_trow + k;
                        g_e1[p].a = SH_OFF[p1_i1] + i;
                        g_e1[p].b = SH_OFF[p1_i2] + j;
                        g_e1[p].v = v;
                        ++p;
                    }
                }
        }
    }

    // ---- TP2: one thread per path (5 valid) ----
    int p2_i1 = -1, p2_i2 = 0, p2_i3 = 0, p2_woff = 0, p2_rowb = 0, p2_want = -1;
    double p2_pw = 0.0;
    if (tid >= 64 && tid < 128) {
        int want = tid - 64;
        int fan2[2] = {0, 0};
        for (int i1 = 0; i1 < 5; ++i1)
            for (int i2 = 0; i2 < 4; ++i2)
                for (int i3 = 0; i3 < 2; ++i3) {
                    int l1 = MID_L[i1], l2 = SH_L[i2], l3 = OUT_L[i3];
                    if (MID_P[i1] * SH_P[i2] == OUT_P[i3] && l3 >= iabs_i(l1 - l2) && l3 <= l1 + l2)
                        fan2[i3] += MID_MUL[i1];
                }
        int idx = 0, woff = 0, rowb = 0;
        for (int i1 = 0; i1 < 5; ++i1)
            for (int i2 = 0; i2 < 4; ++i2)
                for (int i3 = 0; i3 < 2; ++i3) {
                    int l1 = MID_L[i1], l2 = SH_L[i2], l3 = OUT_L[i3];
                    if (!(MID_P[i1] * SH_P[i2] == OUT_P[i3] && l3 >= iabs_i(l1 - l2) && l3 <= l1 + l2)) continue;
                    if (idx == want) {
                        p2_i1 = i1; p2_i2 = i2; p2_i3 = i3; p2_woff = woff; p2_rowb = rowb; p2_want = want;
                        p2_pw = sqrt((double)(2 * l3 + 1) / (double)fan2[i3]);
                    }
                    woff += MID_MUL[i1] * OUT_MUL[i3];
                    rowb += MID_MUL[i1];
                    ++idx;
                }
    }
    int p2_d1 = 0;
    if (p2_i1 >= 0) {
        int l1 = MID_L[p2_i1], l2 = SH_L[p2_i2], l3 = OUT_L[p2_i3];
        p2_d1 = 2 * l1 + 1;
        w3j_real(l1, l2, l3, C);
        int u = MID_MUL[p2_i1], wcnt = OUT_MUL[p2_i3];
        float scale = (float)p2_pw * 0.5f * INV_SQRT2F;  // fold last segsum + graph-sum scale
        for (int uu = 0; uu < u; ++uu) {
            int row = p2_rowb + uu;
            for (int w = 0; w < wcnt; ++w)
                g_M2[row * 8 + OUT_OFF[p2_i3] + w] = scale * w2[p2_woff + uu * wcnt + w];
        }
        int nnz = 0;
        for (int i = 0; i < p2_d1; ++i)
            for (int j = 0; j < p2_d1; ++j)
                if (fabsf(C[(i * 7 + j) * 7 + 0]) > 1e-8f) ++nnz;
        g_p2cnt[p2_want] = u * nnz;
    }
    __syncthreads();
    if (tid == 0) {
        int acc = 0;
        for (int p = 0; p < 8; ++p) { g_p2start[p] = acc; acc += g_p2cnt[p]; }
        g_ne2 = acc;
    }
    __syncthreads();
    if (p2_i1 >= 0) {
        int u = MID_MUL[p2_i1];
        int p = g_p2start[p2_want];
        for (int uu = 0; uu < u; ++uu)
            for (int i = 0; i < p2_d1; ++i)
                for (int j = 0; j < p2_d1; ++j) {
                    float v = C[(i * 7 + j) * 7 + 0];
                    if (fabsf(v) > 1e-8f) {
                        g_e2[p].r = p2_rowb + uu;
                        g_e2[p].a = MID_OFF[p2_i1] + uu * p2_d1 + i;
                        g_e2[p].b = SH_OFF[p2_i2] + j;
                        g_e2[p].v = v;
                        ++p;
                    }
                }
    }
}

// ------------- fused per-cluster kernel: 1 block = 2 clusters (16 nodes) -------------
__global__ __launch_bounds__(256) void fused_kernel(const float* __restrict__ pos,
                                                    const int* __restrict__ esrc,
                                                    const int* __restrict__ edst,
                                                    float* __restrict__ out) {
    __shared__ float shv[112][16];    // SH per edge
    __shared__ float nf1[16][16];     // first aggregated features (incl 1/sqrt2)
    __shared__ float ssum[16][16];    // sum of sh over out-edges per node
    __shared__ float T[16][100];      // dst-summed bilinears (pad 100: conflict-free WMMA A reads)
    __shared__ float NF2[16][NCOLS];  // second aggregated features (GEMM result)
    __shared__ float opart[256][8];

    const int tid = threadIdx.x;
    const int c0 = blockIdx.x * 2;

    // Phase A: spherical harmonics per edge (lmax=3, component norm)
    if (tid < 112) {
        int cl = tid / 56, q = tid % 56;
        int e = (c0 + cl) * 56 + q;
        int s = esrc[e], d = edst[e];
        float x = pos[3 * s + 0] - pos[3 * d + 0];
        float y = pos[3 * s + 1] - pos[3 * d + 1];
        float z = pos[3 * s + 2] - pos[3 * d + 2];
        const float SQ3 = 1.7320508075688772f, SQ5 = 2.2360679774997896f, SQ7 = 2.6457513110645906f;
        const float C56 = 0.91287092917527686f, C38 = 0.61237243569579452f;
        float x2 = x * x, y2 = y * y, z2 = z * z, xz2 = x2 + z2;
        float s20 = SQ3 * x * z, s21 = SQ3 * x * y, s22 = y2 - 0.5f * xz2;
        float s23 = SQ3 * y * z, s24 = 0.5f * SQ3 * (z2 - x2);
        float s30 = C56 * (s20 * z + s24 * x);
        float s31 = SQ5 * s20 * y;
        float s32 = C38 * (4.f * y2 - xz2) * x;
        float s33 = 0.5f * y * (2.f * y2 - 3.f * xz2);
        float s34 = C38 * z * (4.f * y2 - xz2);
        float s35 = SQ5 * s24 * y;
        float s36 = C56 * (s24 * z - s20 * x);
        float* o = shv[tid];
        o[0] = 1.f; o[1] = SQ3 * x; o[2] = SQ3 * y; o[3] = SQ3 * z;
        o[4] = SQ5 * s20; o[5] = SQ5 * s21; o[6] = SQ5 * s22; o[7] = SQ5 * s23; o[8] = SQ5 * s24;
        o[9] = SQ7 * s30; o[10] = SQ7 * s31; o[11] = SQ7 * s32; o[12] = SQ7 * s33;
        o[13] = SQ7 * s34; o[14] = SQ7 * s35; o[15] = SQ7 * s36;
    }
    __syncthreads();

    // Phase A2: nf1 (dst-sum * 1/sqrt2) and ssum (src-sum).  256 thr = 16 nodes x 16 comps
    {
        int n = tid >> 4, c = tid & 15;
        int cl = n >> 3, dl = n & 7;
        float accN = 0.f, accS = 0.f;
        for (int i = 0; i < 8; ++i) {
            if (i == dl) continue;
            int ein  = cl * 56 + i * 7 + (dl - (dl > i));   // edge i -> dl
            int eout = cl * 56 + dl * 7 + (i - (i > dl));   // edge dl -> i
            accN += shv[ein][c];
            accS += shv[eout][c];
        }
        nf1[n][c] = accN * INV_SQRT2F;
        ssum[n][c] = accS;
    }
    __syncthreads();

    // Phase B: T[n][r] = sum over in-edges of sparse Wigner bilinears (row-partitioned)
    {
        int n = tid >> 4, s = tid & 15;
        int cl = n >> 3, dl = n & 7;
        for (int r = s; r < TROWS; r += 16) {
            float acc = 0.f;
            int i0 = g_row_start[r], i1 = g_row_start[r + 1];
            for (int idx = i0; idx < i1; ++idx) {
                Ent en = g_e1[idx];
                float p = 0.f;
                for (int i = 0; i < 8; ++i) {
                    if (i == dl) continue;
                    int ein = cl * 56 + i * 7 + (dl - (dl > i));
                    p += nf1[cl * 8 + i][en.a] * shv[ein][en.b];
                }
                acc += en.v * p;
            }
            T[n][r] = acc;  // rows 86..95 come out 0 (empty ranges) = K padding
        }
    }
    __syncthreads();

    // Phase C: NF2[16][384] = T[16][96] x M1[96][384] via v_wmma_f32_16x16x4_f32
    // 8 waves, each owning 3 N-tiles with 3 accumulators; K-outer loop:
    // A loaded from LDS once per K-step, B via single global_load_b64 per lane from
    // the pre-swizzled g_M1s (loop-invariant VGPR base in AS(1) + immediate offsets),
    // 3 independent WMMAs issued back-to-back.
    {
        int lane = tid & 31, wave = tid >> 5;
        int mrow = lane & 15;     // A: M = lane%16
        int half = lane >> 4;     // A/B: k = r + 2*(lane/16)
        int nbase = wave * 48 + mrow;  // first of 3 col-tiles for this wave
        v8f acc0 = {0.f,0.f,0.f,0.f,0.f,0.f,0.f,0.f};
        v8f acc1 = {0.f,0.f,0.f,0.f,0.f,0.f,0.f,0.f};
        v8f acc2 = {0.f,0.f,0.f,0.f,0.f,0.f,0.f,0.f};
        // one per-lane 64-bit base address; make it opaque so the compiler emits
        // loads with immediate offsets instead of re-materializing the symbol base
        // (s_get_pc chains) for every unrolled load; AS(1) => global_load_b64
        uintptr_t bp_raw = (uintptr_t)((const v2f*)g_M1s + (half * NCOLS + nbase));
        asm("" : "+v"(bp_raw));
        gv2p bp2 = (gv2p)bp_raw;
        for (int kk = 0; kk < 24; ++kk) {
            int kbase = kk * 4 + 2 * half;
            v2f a;
            a.x = T[mrow][kbase + 0];
            a.y = T[mrow][kbase + 1];
            v2f b0 = bp2[kk * 2 * NCOLS +  0];
            v2f b1 = bp2[kk * 2 * NCOLS + 16];
            v2f b2 = bp2[kk * 2 * NCOLS + 32];
            acc0 = __builtin_amdgcn_wmma_f32_16x16x4_f32(false, a, false, b0, (short)0, acc0, false, false);
            acc1 = __builtin_amdgcn_wmma_f32_16x16x4_f32(false, a, false, b1, (short)0, acc1, false, false);
            acc2 = __builtin_amdgcn_wmma_f32_16x16x4_f32(false, a, false, b2, (short)0, acc2, false, false);
        }
        #pragma unroll
        for (int v = 0; v < 8; ++v) {        // D: row m = v + 8*(lane/16), col n = lane%16
            NF2[v + 8 * half][nbase +  0] = acc0[v];
            NF2[v + 8 * half][nbase + 16] = acc1[v];
            NF2[v + 8 * half][nbase + 32] = acc2[v];
        }
    }
    __syncthreads();

    // Phase D: node-level TP2 (collapsed over out-edges) + output reduction
    {
        int n = tid >> 4, s = tid & 15;
        float o[7] = {0.f, 0.f, 0.f, 0.f, 0.f, 0.f, 0.f};
        int ne2 = g_ne2;
        for (int idx = s; idx < ne2; idx += 16) {
            Ent en = g_e2[idx];
            float p = en.v * NF2[n][en.a] * ssum[n][en.b];
            const float* m2 = &g_M2[en.r * 8];
            #pragma unroll
            for (int w = 0; w < 7; ++w) o[w] += m2[w] * p;
        }
        #pragma unroll
        for (int w = 0; w < 7; ++w) opart[tid][w] = o[w];
        opart[tid][7] = 0.f;
    }
    __syncthreads();
    if (tid < 14) {
        int cl = tid / 7, w = tid % 7;
        float acc = 0.f;
        for (int t = cl * 128; t < cl * 128 + 128; ++t) acc += opart[t][w];
        out[(c0 + cl) * 7 + w] = acc;
    }
}

extern "C" void kernel_launch(void* const* d_in, const int* in_sizes, int n_in,
                              void* d_out, int out_size, void* d_ws, size_t ws_size,
                              hipStream_t stream) {
    const float* pos  = (const float*)d_in[0];
    const float* w1   = (const float*)d_in[1];
    const float* w2   = (const float*)d_in[2];
    const int*   esrc = (const int*)d_in[3];
    const int*   edst = (const int*)d_in[4];
    float* out = (float*)d_out;

    int nNodes = in_sizes[0] / 3;   // 32768
    int nClus  = nNodes / 8;        // 4096
    int nBlocks = nClus / 2;        // 2048

    setup_kernel<<<1, 256, 0, stream>>>(w1, w2);
    fused_kernel<<<nBlocks, 256, 0, stream>>>(pos, esrc, edst, out);
}